// AMPS_26001732010118
// MI455X (gfx1250) — compile-verified
//
#include <hip/hip_runtime.h>
#include <math.h>

typedef __attribute__((ext_vector_type(2))) float v2f;
typedef __attribute__((ext_vector_type(8))) float v8f;

#define NS 256   // sites
#define BSZ 2048 // batch

// ---- fp32 WMMA 16x16x4: D = A(16x4) * B(4x16) + C --------------------------
#if __has_builtin(__builtin_amdgcn_wmma_f32_16x16x4_f32)
__device__ __forceinline__ v8f wmma4(v2f a, v2f b, v8f c) {
  return __builtin_amdgcn_wmma_f32_16x16x4_f32(false, a, false, b, (short)0, c,
                                               false, false);
}
#else
__device__ __forceinline__ v8f wmma4(v2f a, v2f b, v8f c) {
  v8f d;
  asm volatile("v_wmma_f32_16x16x4_f32 %0, %1, %2, %3"
               : "=v"(d)
               : "v"(a), "v"(b), "v"(c));
  return d;
}
#endif

// ---------------------------------------------------------------------------
// GEMM orientation:  C'[c,b] = sum_r A'[c,r] * V[b,r],  b = batch (N dim).
// A' column permutation (c = 0..15):  [P0 P1 Q0 Q1 P2 P3 Q2 Q3 | +4 ...]
//   k(c)  = ((c&8)>>1) | ((c&4)>>1) | (c&1),   isQ(c) = (c>>1)&1
//   => C' lanes 0-15:  c[0:1]=P[b,0:1], c[2:3]=Q[b,0:1], c[4:5]=P[b,2:3],
//      c[6:7]=Q[b,2:3]; lanes 16-31 same with k+4.  Combine is two natural
//      v_pk_fma + v_pk_add per tile, no cross-register movs, no shuffles.
// K-slot assignment: WMMA#1 slots -> r={0,1 | 4,5}, WMMA#2 -> r={2,3 | 6,7},
//   so each lane half computes exactly the V components it supplies next step
//   (lanes<16 keep V[0..3], lanes>=16 keep V[4..7]).
// One wave processes 4 batch tiles (64 batch rows), sharing the A operand.
// MODE 0: per-chain contribution -> ws[(n-1)*BSZ + b]  (deterministic)
// MODE 1: atomicAdd into out (fallback if ws too small)
// ---------------------------------------------------------------------------
template <int MODE>
__global__ __launch_bounds__(128) void amps_chain(const float* __restrict__ data,
                                                  const float* __restrict__ Traw,
                                                  float* __restrict__ sink) {
  const int lane = threadIdx.x & 31;
  const int wave = threadIdx.x >> 5;
  const int w    = blockIdx.x * 4 + wave;                       // 0 .. 255*32-1
  const int n    = __builtin_amdgcn_readfirstlane((w >> 5) + 1); // chain 1..255 (SGPR)
  const int grp  = __builtin_amdgcn_readfirstlane(w & 31);       // tile group (SGPR)
  const int col  = lane & 15;               // A'-column c / batch-in-tile
  const bool hiL = lane >= 16;
  const int rb   = hiL ? 4 : 0;             // V-component & K-row base of half
  const int k8   = ((col & 8) >> 1) | ((col & 4) >> 1) | (col & 1); // memory col k
  const bool isQ = ((col >> 1) & 1) != 0;   // (T1-T0) column?
  const int bg   = grp * 64 + col;          // batch row of tile 0

  const v2f* __restrict__ Tp   = (const v2f*)Traw;   // (i=0,i=1) pairs
  const float* __restrict__ dr = data + bg * NS;

  // ---- init at position 0: V[b,k] = d*T[n,0,0,k,0] + (1-d)*T[n,0,0,k,1] + delta_k0
  v2f va[4], vb[4];
  {
    const v2f* p = Tp + n * (NS * 64) + rb;   // (n,0,l=0,k=rb+j,*)
    v2f x0 = p[0], x1 = p[1], x2 = p[2], x3 = p[3];
    const float del = hiL ? 0.0f : 1.0f;
#pragma unroll
    for (int t = 0; t < 4; ++t) {
      const float d = dr[t * 16 * NS];
      const float s = 1.0f - d;
      va[t].x = d * x0.x + s * x0.y + del;
      va[t].y = d * x1.x + s * x1.y;
      vb[t].x = d * x2.x + s * x2.y;
      vb[t].y = d * x3.x + s * x3.y;
    }
  }

  // ---- steps idx = 1 .. n-1 : V <- V + V*T0 + s*(V*(T1-T0))
  for (int idx = 1; idx < n; ++idx) {
    const v2f* p = Tp + (n * NS + idx) * 64 + rb * 8 + k8;  // rows rb+0..3, col k8
    v2f g0 = p[0], g1 = p[8], g2 = p[16], g3 = p[24];
    v2f a0, a1;                                // A' values for this lane (c=col)
    a0.x = isQ ? (g0.y - g0.x) : g0.x;         // r = rb+0
    a0.y = isQ ? (g1.y - g1.x) : g1.x;         // r = rb+1
    a1.x = isQ ? (g2.y - g2.x) : g2.x;         // r = rb+2
    a1.y = isQ ? (g3.y - g3.x) : g3.x;         // r = rb+3

    float sv[4];
#pragma unroll
    for (int t = 0; t < 4; ++t) sv[t] = 1.0f - dr[idx + t * 16 * NS];

    const v8f z = {0.0f, 0.0f, 0.0f, 0.0f, 0.0f, 0.0f, 0.0f, 0.0f};
    v8f cc[4];
#pragma unroll
    for (int t = 0; t < 4; ++t) {
      cc[t] = wmma4(a0, va[t], z);
      cc[t] = wmma4(a1, vb[t], cc[t]);
    }
#pragma unroll
    for (int t = 0; t < 4; ++t) {
      const v2f ss = {sv[t], sv[t]};
      v2f pA = __builtin_shufflevector(cc[t], cc[t], 0, 1);
      v2f qA = __builtin_shufflevector(cc[t], cc[t], 2, 3);
      v2f pB = __builtin_shufflevector(cc[t], cc[t], 4, 5);
      v2f qB = __builtin_shufflevector(cc[t], cc[t], 6, 7);
      va[t] = va[t] + (pA + ss * qA);
      vb[t] = vb[t] + (pB + ss * qB);
    }
  }

  // ---- logits at position n: l_i = sum_l V[b,l]*(T[n,n,l,0,i] + delta_l0)
  {
    const v2f* p = Tp + (n * NS + n) * 64 + rb * 8;  // rows rb+0..3, col 0
    v2f h0 = p[0], h1 = p[8], h2 = p[16], h3 = p[24];
    if (!hiL) { h0.x += 1.0f; h0.y += 1.0f; }        // bias delta at l=0

#pragma unroll
    for (int t = 0; t < 4; ++t) {
      float l0 = va[t].x * h0.x + va[t].y * h1.x + vb[t].x * h2.x + vb[t].y * h3.x;
      float l1 = va[t].x * h0.y + va[t].y * h1.y + vb[t].x * h2.y + vb[t].y * h3.y;
      l0 += __shfl_xor(l0, 16, 32);
      l1 += __shfl_xor(l1, 16, 32);
      float m   = fmaxf(l0, l1);
      float lse = m + logf(expf(l0 - m) + expf(l1 - m));
      float dN  = dr[n + t * 16 * NS];
      float contrib = (l0 - lse) * dN + (l1 - lse) * (1.0f - dN);
      if (!hiL) {
        if (MODE == 0)
          sink[(n - 1) * BSZ + bg + t * 16] = contrib;
        else
          atomicAdd(sink + bg + t * 16, contrib);
      }
    }
  }
}

// out[b] = pos0 contribution + sum over 255 chain contributions (deterministic)
__global__ void amps_reduce(const float* __restrict__ data,
                            const float* __restrict__ Traw,
                            const float* __restrict__ ws,
                            float* __restrict__ out) {
  int b = blockIdx.x * blockDim.x + threadIdx.x;
  if (b >= BSZ) return;
  float t0 = Traw[0] + 1.0f;   // T[0,0,0,0,0] + bias
  float t1 = Traw[1] + 1.0f;
  float m   = fmaxf(t0, t1);
  float lse = m + logf(expf(t0 - m) + expf(t1 - m));
  float d   = data[b * NS];
  float acc = (t0 - lse) * d + (t1 - lse) * (1.0f - d);
  for (int j = 0; j < NS - 1; ++j) acc += ws[j * BSZ + b];
  out[b] = acc;
}

// fallback initializer for the atomic path: out[b] = pos0 contribution
__global__ void amps_pos0(const float* __restrict__ data,
                          const float* __restrict__ Traw,
                          float* __restrict__ out) {
  int b = blockIdx.x * blockDim.x + threadIdx.x;
  if (b >= BSZ) return;
  float t0 = Traw[0] + 1.0f;
  float t1 = Traw[1] + 1.0f;
  float m   = fmaxf(t0, t1);
  float lse = m + logf(expf(t0 - m) + expf(t1 - m));
  float d   = data[b * NS];
  out[b] = (t0 - lse) * d + (t1 - lse) * (1.0f - d);
}

extern "C" void kernel_launch(void* const* d_in, const int* in_sizes, int n_in,
                              void* d_out, int out_size, void* d_ws, size_t ws_size,
                              hipStream_t stream) {
  const float* data = (const float*)d_in[0];   // (2048, 256) f32
  const float* T    = (const float*)d_in[1];   // (256, 256, 8, 8, 2) f32
  float* out        = (float*)d_out;           // (2048,) f32

  const dim3 grid((255 * 32) / 4);             // 4 waves/block, 4 tiles/wave
  const dim3 blk(128);
  const size_t need = (size_t)(NS - 1) * BSZ * sizeof(float);

  if (ws_size >= need) {
    float* ws = (float*)d_ws;
    amps_chain<0><<<grid, blk, 0, stream>>>(data, T, ws);
    amps_reduce<<<(BSZ + 255) / 256, 256, 0, stream>>>(data, T, ws, out);
  } else {
    amps_pos0<<<(BSZ + 255) / 256, 256, 0, stream>>>(data, T, out);
    amps_chain<1><<<grid, blk, 0, stream>>>(data, T, out);
  }
}